// MGNN_IS_66254165508725
// MI455X (gfx1250) — compile-verified
//
#include <hip/hip_runtime.h>
#include <hip/hip_bf16.h>

typedef __attribute__((ext_vector_type(2))) float v2f;
typedef __attribute__((ext_vector_type(8))) float v8f;

__device__ __forceinline__ unsigned ordf(float f) {
    unsigned u = __float_as_uint(f);
    return (u >> 31) ? ~u : (u | 0x80000000u);
}
__device__ __forceinline__ float unordf(unsigned u) {
    return __uint_as_float((u >> 31) ? (u & 0x7fffffffu) : ~u);
}
__device__ __forceinline__ float sigmoidf(float x) {
    return 1.0f / (1.0f + __expf(-x));
}

// ---------------- kernel 0: init accumulators ----------------
__global__ __launch_bounds__(256)
void mg_init(float* den, float* sums, float* deg, unsigned* mmax, int N) {
    long long idx = (long long)blockIdx.x * 256 + threadIdx.x;
    long long tot = (long long)N * 64;
    if (idx < tot) { den[idx] = 0.0f; sums[idx] = 0.0f; mmax[idx] = 0u; }
    if (idx < N)   { deg[idx] = 0.0f; }
}

// ---------------- kernel 1: per-edge WMMA GEMMs ----------------
// wave32 per 16-edge tile; D = E[16x32] @ W[32x(64|64|32)] via chained
// v_wmma_f32_16x16x4_f32 (8 k-steps per 16-col tile).
__global__ __launch_bounds__(256)
void mg_edge_gemm(const float* __restrict__ x,
                  const int*   __restrict__ ei,      // [2][E] (src, tgt)
                  const float* __restrict__ lin1_w, const float* __restrict__ lin1_b,
                  const float* __restrict__ p1_w,   const float* __restrict__ p1_b,
                  const float* __restrict__ p2_w,   const float* __restrict__ p2_b,
                  const float* __restrict__ p_bias1,
                  const float* __restrict__ wa_w,   const float* __restrict__ wa_b,
                  const float* __restrict__ att,
                  float* __restrict__ c_ws, float* __restrict__ wa_ws,
                  float* __restrict__ p_ws, unsigned* __restrict__ mmax,
                  int E) {
    __shared__ float s_p1t[64 * 32];   // transposed: [col][k]
    __shared__ float s_l1t[64 * 32];
    __shared__ float s_wat[32 * 32];
    __shared__ float s_p1b[64], s_l1b[64], s_att[64], s_p2w[64], s_pb1[64];
    __shared__ float s_wab[32];
    __shared__ float s_p2b;

    const int tid = threadIdx.x;
    for (int i = tid; i < 64 * 32; i += 256) {
        int k = i >> 6, c = i & 63;                 // i = k*64 + c (row-major [32][64])
        s_p1t[c * 32 + k] = p1_w[i];
        s_l1t[c * 32 + k] = lin1_w[i];
    }
    for (int i = tid; i < 32 * 32; i += 256) {
        int k = i >> 5, c = i & 31;
        s_wat[c * 32 + k] = wa_w[i];
    }
    if (tid < 64) {
        s_p1b[tid] = p1_b[tid]; s_l1b[tid] = lin1_b[tid];
        s_att[tid] = att[tid];  s_p2w[tid] = p2_w[tid];
        s_pb1[tid] = p_bias1[tid];
    }
    if (tid < 32) s_wab[tid] = wa_b[tid];
    if (tid == 0) s_p2b = p2_b[0];
    __syncthreads();

    const int wave = tid >> 5, lane = tid & 31;
    const long long tile = (long long)blockIdx.x * 8 + wave;
    const long long e0 = tile * 16;
    if (e0 >= E) return;                            // wave-uniform exit

    const int m = lane & 15, half = lane >> 4, n = lane & 15;
    long long e = e0 + m; if (e >= E) e = E - 1;    // clamp (keeps EXEC full)
    const int srcn = ei[e];
    const int tgtn = ei[(long long)E + e];
    const float* xs = x + (long long)srcn * 32;
    const float* xt = x + (long long)tgtn * 32;

    // A fragments: lane l holds e[m][4kb+2h], e[m][4kb+2h+1]
    v2f A[8];
#pragma unroll
    for (int kb = 0; kb < 8; ++kb) {
        int k = kb * 4 + half * 2;
        float2 a = *(const float2*)(xs + k);
        float2 b = *(const float2*)(xt + k);
        A[kb].x = a.x * b.x;
        A[kb].y = a.y * b.y;
    }

    // tgt node id per output row (row mm held by lane mm)
    int tgt_row[8];
#pragma unroll
    for (int r = 0; r < 8; ++r) tgt_row[r] = __shfl(tgtn, r + 8 * half, 32);

    // ---- p1 GEMM (32->64) fused with p2 dot (64->1) ----
    float pdot[8] = {0, 0, 0, 0, 0, 0, 0, 0};
#pragma unroll
    for (int nt = 0; nt < 4; ++nt) {
        v8f acc = {0, 0, 0, 0, 0, 0, 0, 0};
        const int col = nt * 16 + n;
#pragma unroll
        for (int kb = 0; kb < 8; ++kb) {
            int k = kb * 4 + half * 2;
            v2f b; b.x = s_p1t[col * 32 + k]; b.y = s_p1t[col * 32 + k + 1];
            acc = __builtin_amdgcn_wmma_f32_16x16x4_f32(
                false, A[kb], false, b, (short)0, acc, false, false);
        }
        const float bs = s_p1b[col], pw = s_p2w[col], pb1 = s_pb1[col];
#pragma unroll
        for (int r = 0; r < 8; ++r) {
            float h = acc[r] + bs;
            h = (h > 0.0f ? h : 0.0f) + pb1;
            pdot[r] += h * pw;
        }
    }
#pragma unroll
    for (int r = 0; r < 8; ++r) {
        float v = pdot[r];
        v += __shfl_xor(v, 1, 32);
        v += __shfl_xor(v, 2, 32);
        v += __shfl_xor(v, 4, 32);
        v += __shfl_xor(v, 8, 32);
        pdot[r] = v;                                // full row-sum on n==0 lanes
    }
    if (n == 0) {
#pragma unroll
        for (int r = 0; r < 8; ++r) {
            long long ee = e0 + r + 8 * half;
            if (ee < E) p_ws[ee] = sigmoidf(pdot[r] + s_p2b);
        }
    }

    // ---- lin1 GEMM (32->64) -> c = leaky(att * relu(.)) + segment max ----
#pragma unroll
    for (int nt = 0; nt < 4; ++nt) {
        v8f acc = {0, 0, 0, 0, 0, 0, 0, 0};
        const int col = nt * 16 + n;
#pragma unroll
        for (int kb = 0; kb < 8; ++kb) {
            int k = kb * 4 + half * 2;
            v2f b; b.x = s_l1t[col * 32 + k]; b.y = s_l1t[col * 32 + k + 1];
            acc = __builtin_amdgcn_wmma_f32_16x16x4_f32(
                false, A[kb], false, b, (short)0, acc, false, false);
        }
        const float bs = s_l1b[col], av = s_att[col];
#pragma unroll
        for (int r = 0; r < 8; ++r) {
            float t = acc[r] + bs;
            t = t > 0.0f ? t : 0.0f;
            float cc = av * t;
            cc = cc > 0.0f ? cc : 0.01f * cc;       // leaky_relu default slope
            long long ee = e0 + r + 8 * half;
            if (ee < E) {
                c_ws[ee * 64 + col] = cc;
                atomicMax(&mmax[(long long)tgt_row[r] * 64 + col], ordf(cc));
            }
        }
    }

    // ---- wa GEMM (32->32) ----
#pragma unroll
    for (int nt = 0; nt < 2; ++nt) {
        v8f acc = {0, 0, 0, 0, 0, 0, 0, 0};
        const int col = nt * 16 + n;
#pragma unroll
        for (int kb = 0; kb < 8; ++kb) {
            int k = kb * 4 + half * 2;
            v2f b; b.x = s_wat[col * 32 + k]; b.y = s_wat[col * 32 + k + 1];
            acc = __builtin_amdgcn_wmma_f32_16x16x4_f32(
                false, A[kb], false, b, (short)0, acc, false, false);
        }
        const float bs = s_wab[col];
#pragma unroll
        for (int r = 0; r < 8; ++r) {
            long long ee = e0 + r + 8 * half;
            if (ee < E) wa_ws[ee * 32 + col] = acc[r] + bs;
        }
    }
}

// ---------------- kernel 2: deterministic per-graph top-k mask ----------------
__global__ __launch_bounds__(256)
void mg_topk(const float* __restrict__ p_ws, float* __restrict__ pm_ws,
             const int* __restrict__ num_graphs, const int* __restrict__ layer,
             int E) {
    __shared__ float sp[2048];
    const int B = num_graphs[0];
    const int L = layer[0];
    const int EPG = E / (B > 0 ? B : 1);
    const int k = (int)floorf(0.5f * (float)EPG);   // TOPK = 0.5
    const bool useL = (EPG <= 2048);
    for (int g = blockIdx.x; g < B; g += gridDim.x) {
        const float* pg = p_ws + (long long)g * EPG;
        float* og = pm_ws + (long long)g * EPG;
        if (L <= 0) {
            for (int j = threadIdx.x; j < EPG; j += 256) og[j] = pg[j];
            continue;
        }
        if (useL)
            for (int j = threadIdx.x; j < EPG; j += 256) sp[j] = pg[j];
        __syncthreads();
        for (int j = threadIdx.x; j < EPG; j += 256) {
            const float pj = useL ? sp[j] : pg[j];
            int rank = 0;
            for (int i = 0; i < EPG; ++i) {
                const float pi = useL ? sp[i] : pg[i];
                rank += (pi > pj) || (pi == pj && i < j);  // stable desc order
            }
            og[j] = (rank < k) ? pj : 0.0f;
        }
        __syncthreads();
    }
}

// ---------------- kernel 3: softmax denominator + degree ----------------
__global__ __launch_bounds__(256)
void mg_denom(const float* __restrict__ c_ws, const int* __restrict__ ei,
              const unsigned* __restrict__ mmax, float* __restrict__ den,
              float* __restrict__ deg, int E) {
    long long idx = (long long)blockIdx.x * 256 + threadIdx.x;
    if (idx >= (long long)E * 64) return;
    long long e = idx >> 6; int col = (int)(idx & 63);
    int tgt = ei[(long long)E + e];
    float mv = unordf(mmax[(long long)tgt * 64 + col]);
    float ex = __expf(c_ws[idx] - mv);
    atomicAdd(&den[(long long)tgt * 64 + col], ex);
    if (col == 0) atomicAdd(&deg[tgt], 1.0f);
}

// ---------------- kernel 4: alpha + message + scatter-add ----------------
// one wave per (edge, head); lane = feature dim d
__global__ __launch_bounds__(256)
void mg_message(const float* __restrict__ c_ws, const float* __restrict__ wa_ws,
                const float* __restrict__ pm_ws, const int* __restrict__ ei,
                const unsigned* __restrict__ mmax, const float* __restrict__ den,
                float* __restrict__ sums, int E) {
    long long gw = (long long)blockIdx.x * 8 + (threadIdx.x >> 5);
    if (gw >= (long long)E * 2) return;
    long long e = gw >> 1;
    int h = (int)(gw & 1);
    int d = threadIdx.x & 31;
    int col = h * 32 + d;
    int tgt = ei[(long long)E + e];
    float mv = unordf(mmax[(long long)tgt * 64 + col]);
    float ex = __expf(c_ws[e * 64 + col] - mv);
    float v = ex / (den[(long long)tgt * 64 + col] + 1e-16f);
    float alpha = v;
    alpha += __shfl_xor(alpha, 1, 32);
    alpha += __shfl_xor(alpha, 2, 32);
    alpha += __shfl_xor(alpha, 4, 32);
    alpha += __shfl_xor(alpha, 8, 32);
    alpha += __shfl_xor(alpha, 16, 32);
    float w  = wa_ws[e * 32 + d];
    float pv = pm_ws[e];
    float msg = sigmoidf(w * alpha * pv);
    atomicAdd(&sums[(long long)tgt * 64 + col], msg);
}

// ---------------- kernel 5: mean + bias ----------------
__global__ __launch_bounds__(256)
void mg_finalize(const float* __restrict__ sums, const float* __restrict__ deg,
                 const float* __restrict__ bias, float* __restrict__ out, int N) {
    long long idx = (long long)blockIdx.x * 256 + threadIdx.x;
    if (idx >= (long long)N * 64) return;
    int nn = (int)(idx >> 6), col = (int)(idx & 63);
    out[idx] = sums[idx] / fmaxf(deg[nn], 1.0f) + bias[col];
}

extern "C" void kernel_launch(void* const* d_in, const int* in_sizes, int n_in,
                              void* d_out, int out_size, void* d_ws, size_t ws_size,
                              hipStream_t stream) {
    const float* x       = (const float*)d_in[0];
    const float* lin1_w  = (const float*)d_in[1];
    const float* lin1_b  = (const float*)d_in[2];
    const float* p1_w    = (const float*)d_in[3];
    const float* p1_b    = (const float*)d_in[4];
    const float* p2_w    = (const float*)d_in[5];
    const float* p2_b    = (const float*)d_in[6];
    const float* p_bias1 = (const float*)d_in[7];
    const float* wa_w    = (const float*)d_in[9];
    const float* wa_b    = (const float*)d_in[10];
    const float* att     = (const float*)d_in[11];
    const float* bias    = (const float*)d_in[12];
    const int*   ei      = (const int*)d_in[13];
    const int*   ngr     = (const int*)d_in[15];
    const int*   layer   = (const int*)d_in[16];
    float* out = (float*)d_out;

    const int N = in_sizes[0] / 32;
    const int E = in_sizes[13] / 2;

    // workspace layout (256B aligned slices)
    char* base = (char*)d_ws;
    size_t off = 0;
    auto alloc = [&](size_t bytes) {
        char* p = base + off;
        off += (bytes + 255) & ~(size_t)255;
        return p;
    };
    float*    c_ws  = (float*)alloc((size_t)E * 64 * 4);
    float*    wa_ws = (float*)alloc((size_t)E * 32 * 4);
    float*    p_ws  = (float*)alloc((size_t)E * 4);
    float*    pm_ws = (float*)alloc((size_t)E * 4);
    unsigned* mmax  = (unsigned*)alloc((size_t)N * 64 * 4);
    float*    den   = (float*)alloc((size_t)N * 64 * 4);
    float*    sums  = (float*)alloc((size_t)N * 64 * 4);
    float*    deg   = (float*)alloc((size_t)N * 4);
    (void)ws_size;

    long long nwords = (long long)N * 64;
    mg_init<<<(unsigned)((nwords + 255) / 256), 256, 0, stream>>>(den, sums, deg, mmax, N);

    long long tiles = ((long long)E + 15) / 16;
    mg_edge_gemm<<<(unsigned)((tiles + 7) / 8), 256, 0, stream>>>(
        x, ei, lin1_w, lin1_b, p1_w, p1_b, p2_w, p2_b, p_bias1,
        wa_w, wa_b, att, c_ws, wa_ws, p_ws, mmax, E);

    mg_topk<<<1024, 256, 0, stream>>>(p_ws, pm_ws, ngr, layer, E);

    long long ecols = (long long)E * 64;
    mg_denom<<<(unsigned)((ecols + 255) / 256), 256, 0, stream>>>(c_ws, ei, mmax, den, deg, E);

    long long ewaves = (long long)E * 2;
    mg_message<<<(unsigned)((ewaves + 7) / 8), 256, 0, stream>>>(
        c_ws, wa_ws, pm_ws, ei, mmax, den, sums, E);

    mg_finalize<<<(unsigned)((nwords + 255) / 256), 256, 0, stream>>>(sums, deg, bias, out, N);
}